// DGLJTNNEncoder_10144712753478
// MI455X (gfx1250) — compile-verified
//
#include <hip/hip_runtime.h>
#include <hip/hip_bf16.h>
#include <math.h>

#define HID 256
#define NT_NODES 31
#define EDGES_PER_TREE 30
#define N_TREES 4096
#define N_NODES (N_TREES * NT_NODES)       // 126976
#define NE_HALF (N_TREES * EDGES_PER_TREE) // 122880
#define M_BLK 64                           // edges per block
#define KCHUNK 32                          // K depth staged in LDS per step
#define MPAD (HID + 8)                     // padded m-tile row (bank-conflict free)

typedef float v2f __attribute__((ext_vector_type(2)));
typedef float v8f __attribute__((ext_vector_type(8)));

// fp32 WMMA: D(16x16,f32) = A(16x4,f32) * B(4x16,f32) + C
__device__ __forceinline__ v8f wmma4(v2f a, v2f b, v8f c) {
  return __builtin_amdgcn_wmma_f32_16x16x4_f32(false, a, false, b, (short)0, c,
                                               false, false);
}

__device__ __forceinline__ float sigmoidf_(float v) {
  return 1.0f / (1.0f + __expf(-v));
}

// CDNA5 async copy: 16 bytes global -> LDS, tracked by ASYNCcnt.
__device__ __forceinline__ void async_copy_b128(unsigned lds_off,
                                                const float* gsrc) {
  asm volatile("global_load_async_to_lds_b128 %0, %1, off" ::"v"(lds_off),
               "v"(gsrc)
               : "memory");
}
__device__ __forceinline__ void wait_async0() {
  asm volatile("s_wait_asynccnt 0x0" ::: "memory");
}

// Stage one packed-weight K-chunk (KCHUNK x 256 floats, contiguous in the
// packed layout) into LDS with the async engine. 256 threads, 8 x b128 each.
__device__ __forceinline__ void stage_chunk(float* buf, const float* gsrc,
                                            int tid) {
  unsigned base = (unsigned)(uintptr_t)buf;  // low 32 bits = LDS offset
#pragma unroll
  for (int i = 0; i < (KCHUNK * HID) / (4 * 256); ++i) {  // 8 iterations
    int idx = tid + i * 256;
    async_copy_b128(base + idx * 16, gsrc + idx * 4);
  }
}

// ---------------------------------------------------------------------------
// x = emb[wid]; s = rm = 0
// ---------------------------------------------------------------------------
__global__ void gather_zero_kernel(const int* __restrict__ wid,
                                   const float* __restrict__ emb,
                                   float* __restrict__ x, float* __restrict__ s,
                                   float* __restrict__ rm) {
  int n = blockIdx.x;
  int c = threadIdx.x;
  int w = wid[n];
  size_t o = (size_t)n * HID + c;
  x[o] = emb[(size_t)w * HID + c];
  s[o] = 0.0f;
  rm[o] = 0.0f;
}

__global__ void zero_sr_kernel(float* __restrict__ s, float* __restrict__ rm) {
  size_t i = (size_t)blockIdx.x * blockDim.x + threadIdx.x;
  s[i] = 0.0f;
  rm[i] = 0.0f;
}

// ---------------------------------------------------------------------------
// Pack row-major W[K][256] -> (k-pair, n, k&1) so a B fragment is one 8B load:
//   Wp[(k>>1)*512 + n*2 + (k&1)] = W[k*256 + n]
// A K-chunk [kc,kc+KCHUNK) is contiguous: floats [kc*256, (kc+KCHUNK)*256).
// ---------------------------------------------------------------------------
__global__ void pack_w_kernel(const float* __restrict__ W,
                              float* __restrict__ Wp) {
  int k = blockIdx.x;
  int n = threadIdx.x;
  Wp[(size_t)(k >> 1) * 512 + n * 2 + (k & 1)] = W[(size_t)k * HID + n];
}

// ---------------------------------------------------------------------------
// One tree level of GRU messages. Block = 256 threads (8 waves), 64 edges.
// Wave layout: mt = wave&3 -> 16-row M subtile; nh = wave>>2 -> 128-col half.
// Weights are staged into LDS in K-chunks via async-to-LDS; B fragments are
// ds_load_b64, A fragments global/LDS b64 loads.
//   z = sigmoid([x_s,s_s]@Wz+bz); h = tanh([x_s,rm_s]@Wh+bh)
//   m = (1-z)*s_s + z*h;  r = sigmoid(x_d@Wr + m@Ur + bu)
// UP:  s/rm[parent] = sibling-pair sums (pairs are even-aligned, no atomics)
// DOWN: s/rm[child] = m / m*r
// ---------------------------------------------------------------------------
template <bool UP>
__global__ __launch_bounds__(256) void level_msg_kernel(
    const float* __restrict__ x, float* __restrict__ s, float* __restrict__ rm,
    const float* __restrict__ WzP, const float* __restrict__ bz,
    const float* __restrict__ WhP, const float* __restrict__ bh,
    const float* __restrict__ WrP, const float* __restrict__ UrP,
    const float* __restrict__ bu, float* __restrict__ m_out, int CB, int EPT) {
  __shared__ float wbufA[KCHUNK * HID];  // packed weight chunk (Wz / Wr)
  __shared__ float wbufB[KCHUNK * HID];  // packed weight chunk (Wh / Ur)
  __shared__ float m_tile[M_BLK][MPAD];
  __shared__ int srcL[M_BLK], dstL[M_BLK], eidL[M_BLK];

  const int tid = threadIdx.x;
  if (tid < M_BLK) {
    int e = blockIdx.x * M_BLK + tid;
    int tree = e / EPT;
    int j = e - tree * EPT;
    int chl = CB + j;  // local child id
    int child = tree * NT_NODES + chl;
    int par = tree * NT_NODES + ((chl - 1) >> 1);
    srcL[tid] = UP ? child : par;
    dstL[tid] = UP ? par : child;
    eidL[tid] = tree * EDGES_PER_TREE + (chl - 1);
  }
  __syncthreads();

  const int lane = tid & 31;
  const int wave = tid >> 5;
  const int half = lane >> 4;  // lanes 16..31: K+2 / M+8
  const int l16 = lane & 15;
  const int mt = wave & 3;
  const int rbase = mt * 16;
  const int wcol = (wave >> 2) * 128;  // 8 N-tiles of 16

  const float* xsrc = x + (size_t)srcL[rbase + l16] * HID;
  const float* ssrc = s + (size_t)srcL[rbase + l16] * HID;
  const float* rmsrc = rm + (size_t)srcL[rbase + l16] * HID;

  v8f accz[8] = {};
  v8f acch[8] = {};

  // ---- GEMM 1a: shared A = x_src, K in [0,256) ----
  for (int c = 0; c < HID / KCHUNK; ++c) {
    __syncthreads();
    stage_chunk(wbufA, WzP + (size_t)(c * KCHUNK) * HID, tid);
    stage_chunk(wbufB, WhP + (size_t)(c * KCHUNK) * HID, tid);
    wait_async0();
    __syncthreads();
    for (int ks = 0; ks < KCHUNK; ks += 4) {
      int ka = ks + 2 * half;
      v2f a = *(const v2f*)(xsrc + c * KCHUNK + ka);
      const float* bzp = wbufA + (ka >> 1) * 512;
      const float* bhp = wbufB + (ka >> 1) * 512;
#pragma unroll
      for (int nt = 0; nt < 8; ++nt) {
        int n2 = (wcol + nt * 16 + l16) * 2;
        accz[nt] = wmma4(a, *(const v2f*)(bzp + n2), accz[nt]);
        acch[nt] = wmma4(a, *(const v2f*)(bhp + n2), acch[nt]);
      }
    }
  }
  // ---- GEMM 1b: A_z = s_src, A_h = rm_src, K in [256,512) ----
  for (int c = 0; c < HID / KCHUNK; ++c) {
    __syncthreads();
    stage_chunk(wbufA, WzP + (size_t)(HID + c * KCHUNK) * HID, tid);
    stage_chunk(wbufB, WhP + (size_t)(HID + c * KCHUNK) * HID, tid);
    wait_async0();
    __syncthreads();
    for (int ks = 0; ks < KCHUNK; ks += 4) {
      int ka = ks + 2 * half;
      v2f az = *(const v2f*)(ssrc + c * KCHUNK + ka);
      v2f ah = *(const v2f*)(rmsrc + c * KCHUNK + ka);
      const float* bzp = wbufA + (ka >> 1) * 512;
      const float* bhp = wbufB + (ka >> 1) * 512;
#pragma unroll
      for (int nt = 0; nt < 8; ++nt) {
        int n2 = (wcol + nt * 16 + l16) * 2;
        accz[nt] = wmma4(az, *(const v2f*)(bzp + n2), accz[nt]);
        acch[nt] = wmma4(ah, *(const v2f*)(bhp + n2), acch[nt]);
      }
    }
  }

  // ---- epilogue 1: m = (1-z)*s_s + z*h ; stage in LDS + write m_out ----
#pragma unroll
  for (int nt = 0; nt < 8; ++nt) {
    int col = wcol + nt * 16 + l16;
    float bzc = bz[col];
    float bhc = bh[col];
#pragma unroll
    for (int v = 0; v < 8; ++v) {
      int row = rbase + v + 8 * half;
      float zv = sigmoidf_(accz[nt][v] + bzc);
      float hv = tanhf(acch[nt][v] + bhc);
      float ssv = s[(size_t)srcL[row] * HID + col];
      float mv = (1.0f - zv) * ssv + zv * hv;
      m_tile[row][col] = mv;
      m_out[(size_t)eidL[row] * HID + col] = mv;
    }
  }

  // ---- GEMM 2: r-logit = x_dst @ Wr + m @ Ur (A_m from padded LDS tile) ----
  const float* xdst = x + (size_t)dstL[rbase + l16] * HID;
  const float* mrow = &m_tile[rbase + l16][0];
  v8f accr[8] = {};
  for (int c = 0; c < HID / KCHUNK; ++c) {
    __syncthreads();  // also fences m_tile writes / prior buffer reads
    stage_chunk(wbufA, WrP + (size_t)(c * KCHUNK) * HID, tid);
    stage_chunk(wbufB, UrP + (size_t)(c * KCHUNK) * HID, tid);
    wait_async0();
    __syncthreads();
    for (int ks = 0; ks < KCHUNK; ks += 4) {
      int ka = ks + 2 * half;
      v2f axd = *(const v2f*)(xdst + c * KCHUNK + ka);
      v2f am = *(const v2f*)(mrow + c * KCHUNK + ka);
      const float* brp = wbufA + (ka >> 1) * 512;
      const float* bup = wbufB + (ka >> 1) * 512;
#pragma unroll
      for (int nt = 0; nt < 8; ++nt) {
        int n2 = (wcol + nt * 16 + l16) * 2;
        accr[nt] = wmma4(axd, *(const v2f*)(brp + n2), accr[nt]);
        accr[nt] = wmma4(am, *(const v2f*)(bup + n2), accr[nt]);
      }
    }
  }

  // ---- epilogue 2: scatter s/rm to destinations ----
#pragma unroll
  for (int nt = 0; nt < 8; ++nt) {
    int col = wcol + nt * 16 + l16;
    float buc = bu[col];
    float mv[8], mrv[8];
#pragma unroll
    for (int v = 0; v < 8; ++v) {
      int row = rbase + v + 8 * half;
      float rv = sigmoidf_(accr[nt][v] + buc);
      float m_ = m_tile[row][col];
      mv[v] = m_;
      mrv[v] = m_ * rv;
    }
    if (UP) {
      // sibling pair (rows v,v+1) share one parent -> plain pair-sum store
#pragma unroll
      for (int v = 0; v < 8; v += 2) {
        int row = rbase + v + 8 * half;
        int d = dstL[row];
        s[(size_t)d * HID + col] = mv[v] + mv[v + 1];
        rm[(size_t)d * HID + col] = mrv[v] + mrv[v + 1];
      }
    } else {
#pragma unroll
      for (int v = 0; v < 8; ++v) {
        int row = rbase + v + 8 * half;
        int d = dstL[row];
        s[(size_t)d * HID + col] = mv[v];
        rm[(size_t)d * HID + col] = mrv[v];
      }
    }
  }
}

// ---------------------------------------------------------------------------
// h_root = relu([x[root], m_up[e0]+m_up[e0+1]] @ Wf + bf); 16 trees per block.
// ---------------------------------------------------------------------------
__global__ __launch_bounds__(128) void root_kernel(
    const float* __restrict__ x, const float* __restrict__ m_up,
    const float* __restrict__ WfP, const float* __restrict__ bf,
    float* __restrict__ h_root) {
  __shared__ float acc_tile[16][MPAD];
  const int tid = threadIdx.x;
  for (int i = tid; i < 16 * HID; i += 128) {
    int row = i >> 8;
    int col = i & 255;
    int tree = blockIdx.x * 16 + row;
    size_t e0 = (size_t)tree * EDGES_PER_TREE * HID;
    acc_tile[row][col] = m_up[e0 + col] + m_up[e0 + HID + col];
  }
  __syncthreads();

  const int lane = tid & 31;
  const int wave = tid >> 5;
  const int half = lane >> 4;
  const int l16 = lane & 15;
  const int wcol = wave * 64;

  int tree_row = blockIdx.x * 16 + l16;
  const float* xr = x + (size_t)tree_row * NT_NODES * HID;  // root = tree*31
  const float* arow = &acc_tile[l16][0];

  v8f acc[4] = {};
  for (int kk = 0; kk < HID; kk += 4) {
    int ka = kk + 2 * half;
    v2f ax = *(const v2f*)(xr + ka);
    v2f aa = *(const v2f*)(arow + ka);
    size_t kp1 = (size_t)(ka >> 1) * 512;
    size_t kp2 = (size_t)((ka + HID) >> 1) * 512;
#pragma unroll
    for (int nt = 0; nt < 4; ++nt) {
      int n2 = (wcol + nt * 16 + l16) * 2;
      v2f b1 = *(const v2f*)(WfP + kp1 + n2);
      v2f b2 = *(const v2f*)(WfP + kp2 + n2);
      acc[nt] = wmma4(ax, b1, acc[nt]);
      acc[nt] = wmma4(aa, b2, acc[nt]);
    }
  }
#pragma unroll
  for (int nt = 0; nt < 4; ++nt) {
    int col = wcol + nt * 16 + l16;
    float bfc = bf[col];
#pragma unroll
    for (int v = 0; v < 8; ++v) {
      int tr = blockIdx.x * 16 + v + 8 * half;
      h_root[(size_t)tr * HID + col] = fmaxf(acc[nt][v] + bfc, 0.0f);
    }
  }
}

// ---------------------------------------------------------------------------
extern "C" void kernel_launch(void* const* d_in, const int* in_sizes, int n_in,
                              void* d_out, int out_size, void* d_ws,
                              size_t ws_size, hipStream_t stream) {
  (void)in_sizes; (void)n_in; (void)out_size; (void)ws_size;
  const int* wid = (const int*)d_in[0];
  const float* emb = (const float*)d_in[1];
  const float* Wz = (const float*)d_in[2];
  const float* bz = (const float*)d_in[3];
  const float* Wh = (const float*)d_in[4];
  const float* bh = (const float*)d_in[5];
  const float* Wr = (const float*)d_in[6];
  const float* Ur = (const float*)d_in[7];
  const float* bu = (const float*)d_in[8];
  const float* Wf = (const float*)d_in[9];
  const float* bf = (const float*)d_in[10];

  float* out = (float*)d_out;
  float* m_up = out;
  float* m_down = out + (size_t)NE_HALF * HID;
  float* h_root = out + (size_t)2 * NE_HALF * HID;

  float* ws = (float*)d_ws;
  float* x = ws;
  float* s = x + (size_t)N_NODES * HID;
  float* rm = s + (size_t)N_NODES * HID;
  float* WzP = rm + (size_t)N_NODES * HID;
  float* WhP = WzP + (size_t)512 * HID;
  float* WrP = WhP + (size_t)512 * HID;
  float* UrP = WrP + (size_t)256 * HID;
  float* WfP = UrP + (size_t)256 * HID;

  pack_w_kernel<<<512, 256, 0, stream>>>(Wz, WzP);
  pack_w_kernel<<<512, 256, 0, stream>>>(Wh, WhP);
  pack_w_kernel<<<256, 256, 0, stream>>>(Wr, WrP);
  pack_w_kernel<<<256, 256, 0, stream>>>(Ur, UrP);
  pack_w_kernel<<<512, 256, 0, stream>>>(Wf, WfP);
  gather_zero_kernel<<<N_NODES, 256, 0, stream>>>(wid, emb, x, s, rm);

  // bottom-up: deepest level first (children depth d+1 -> parents depth d)
  for (int d = 3; d >= 0; --d) {
    int EPT = 2 << d;  // children per tree at this level
    int CB = EPT - 1;  // first local child id
    int nE = N_TREES * EPT;
    level_msg_kernel<true><<<nE / M_BLK, 256, 0, stream>>>(
        x, s, rm, WzP, bz, WhP, bh, WrP, UrP, bu, m_up, CB, EPT);
  }
  zero_sr_kernel<<<(N_NODES * HID) / 256, 256, 0, stream>>>(s, rm);
  // top-down
  for (int d = 0; d < 4; ++d) {
    int EPT = 2 << d;
    int CB = EPT - 1;
    int nE = N_TREES * EPT;
    level_msg_kernel<false><<<nE / M_BLK, 256, 0, stream>>>(
        x, s, rm, WzP, bz, WhP, bh, WrP, UrP, bu, m_down, CB, EPT);
  }
  root_kernel<<<N_TREES / 16, 128, 0, stream>>>(x, m_up, WfP, bf, h_root);
}